// TanhNewtonImplicitLayer_13932873908269
// MI455X (gfx1250) — compile-verified
//
#include <hip/hip_runtime.h>
#include <hip/hip_bf16.h>
#include <math.h>

// Fixed point of z = tanh(W z + x), B=512 rows x D=256, fp32.
// Picard iteration with the z@W^T GEMM on V_WMMA_F32_16X16X4_F32.

typedef __attribute__((ext_vector_type(2))) float v2f;
typedef __attribute__((ext_vector_type(8))) float v8f;

#define D_DIM 256
#define B_DIM 512
#define ZS    260      // padded LDS row stride (floats); 260 % 64 == 4 -> 16 rows hit 16 distinct banks
#define ITERS 50       // Picard steps; contraction ~0.5 => ~1e-15, far under fp32 noise

// Branch-free tanh: tanh(v) = 1 - 2/(exp(2v)+1).
// v_mul + v_exp_f32 + v_add + v_rcp_f32 + v_fma; exact at +/-inf, no EXEC-masked
// regions (keeps EXEC all-ones through the WMMA loop). ~1 ulp hw trans ops =>
// ~1e-7 abs error, same order as the reference's 1e-4 Frobenius stop slack.
__device__ __forceinline__ float fast_tanh(float v) {
    float e = __expf(2.0f * v);                    // v_exp_f32
    float r = __builtin_amdgcn_rcpf(e + 1.0f);     // v_rcp_f32
    return fmaf(-2.0f, r, 1.0f);
}

__global__ __launch_bounds__(512, 1)
void tanh_fixed_point_wmma(const float* __restrict__ x,
                           const float* __restrict__ W,
                           float* __restrict__ out)
{
    __shared__ float zsh[16 * ZS];   // current z tile, 16 rows x 256 cols (padded)

    const int tid  = threadIdx.x;
    const int wave = tid >> 5;        // 16 waves: one 16-column output tile each
    const int lane = tid & 31;
    const int nIdx = lane & 15;       // N within tile (B/C/D layouts) or M row (A layout)
    const int half = lane >> 4;       // 0 = lanes 0-15, 1 = lanes 16-31
    const int n0   = wave * 16;       // column-tile base
    const int rowBase = blockIdx.x * 16;

    // ---- x fragment held in C/D-matrix layout: elem i <-> row M = i + 8*half, col n0+nIdx
    float xf[8];
#pragma unroll
    for (int i = 0; i < 8; ++i) {
        const int M = i + 8 * half;
        xf[i] = x[(rowBase + M) * D_DIM + n0 + nIdx];
    }

    // ---- warm this wave's W panel (rows n0..n0+15, 1KB each) into the cache hierarchy
    {
        const float* wrow = &W[(n0 + nIdx) * D_DIM];
#pragma unroll
        for (int j = 0; j < 4; ++j)
            __builtin_prefetch(wrow + half * 128 + j * 32, 0, 3);  // global_prefetch_b8
    }

    // ---- z0 = tanh(x), stored to LDS in C-layout addressing
#pragma unroll
    for (int i = 0; i < 8; ++i) {
        const int M = i + 8 * half;
        zsh[M * ZS + n0 + nIdx] = fast_tanh(xf[i]);
    }
    __syncthreads();

    // A-fragment addressing (16x4 f32 A layout): lanes 0-15 hold K={k0,k0+1},
    // lanes 16-31 hold K={k0+2,k0+3}, row M = lane&15.
    const int aRow = nIdx;
    const int kOff = 2 * half;

    for (int it = 0; it <= ITERS; ++it) {           // last pass (it==ITERS) IS the output expr
        v8f c = {};                                  // accumulate z @ W^T, add x afterwards
#pragma unroll 8
        for (int kk = 0; kk < D_DIM / 4; ++kk) {
            const int k0 = kk * 4;
            v2f a, b;
            // A = z[16 x 4] from LDS (conflict-free via padded stride)
            a.x = zsh[aRow * ZS + k0 + kOff];
            a.y = zsh[aRow * ZS + k0 + kOff + 1];
            // B = (W^T)[4 x 16]: B[k][n] = W[n0+n][k]; per-lane 8B contiguous load
            const float* wp = &W[(n0 + nIdx) * D_DIM + k0 + kOff];
            b.x = wp[0];
            b.y = wp[1];
            c = __builtin_amdgcn_wmma_f32_16x16x4_f32(false, a, false, b,
                                                      (short)0, c, false, false);
        }

        float zn[8];
#pragma unroll
        for (int i = 0; i < 8; ++i)
            zn[i] = fast_tanh(c[i] + xf[i]);         // TRANS ops, co-execute w/ matrix pipe

        if (it == ITERS) {                           // uniform branch: EXEC stays all-ones
#pragma unroll
            for (int i = 0; i < 8; ++i) {
                const int M = i + 8 * half;
                out[(rowBase + M) * D_DIM + n0 + nIdx] = zn[i];
            }
        } else {
            __syncthreads();                         // all waves done reading zsh
#pragma unroll
            for (int i = 0; i < 8; ++i) {
                const int M = i + 8 * half;
                zsh[M * ZS + n0 + nIdx] = zn[i];
            }
            __syncthreads();                         // writes visible to all waves
        }
    }
}

extern "C" void kernel_launch(void* const* d_in, const int* in_sizes, int n_in,
                              void* d_out, int out_size, void* d_ws, size_t ws_size,
                              hipStream_t stream) {
    const float* x = (const float*)d_in[0];   // [512, 256]
    const float* W = (const float*)d_in[1];   // [256, 256]
    float* out = (float*)d_out;               // [512, 256]
    (void)in_sizes; (void)n_in; (void)out_size; (void)d_ws; (void)ws_size;

    dim3 grid(B_DIM / 16);    // 32 row tiles
    dim3 block(512);          // 16 waves, one 16x16 output tile each
    hipLaunchKernelGGL(tanh_fixed_point_wmma, grid, block, 0, stream, x, W, out);
}